// MPN_55310588838496
// MI455X (gfx1250) — compile-verified
//
#include <hip/hip_runtime.h>

typedef float v2f __attribute__((ext_vector_type(2)));
typedef float v8f __attribute__((ext_vector_type(8)));

#define XCOLS 36   // 2*NF + 4
#define NFD   16
#define HD    128

// ---------------- degree / norm prep ----------------

__global__ __launch_bounds__(256) void degree_kernel(
    const long long* __restrict__ ei, float* __restrict__ deg, long long E) {
  long long e = (long long)blockIdx.x * blockDim.x + threadIdx.x;
  if (e >= E) return;
  // doubled edge list: cols are ei1 (first half) and ei0 (second half)
  atomicAdd(&deg[ei[E + e]], 1.0f);
  atomicAdd(&deg[ei[e]], 1.0f);
}

__global__ __launch_bounds__(256) void dis_kernel(
    const float* __restrict__ deg, float* __restrict__ dis, int n) {
  int i = blockIdx.x * blockDim.x + threadIdx.x;
  if (i >= n) return;
  float d = deg[i];
  dis[i] = (d > 0.0f) ? rsqrtf(d) : 0.0f;
}

__global__ __launch_bounds__(256) void norm_kernel(
    const long long* __restrict__ ei, const float* __restrict__ dis,
    float* __restrict__ nrm, long long E) {
  long long e = (long long)blockIdx.x * blockDim.x + threadIdx.x;
  if (e >= 2 * E) return;
  long long r, c;
  if (e < E) { r = ei[e];           c = ei[E + e]; }
  else       { r = ei[E + (e - E)]; c = ei[e - E]; }
  nrm[e] = dis[r] * dis[c];
}

__global__ __launch_bounds__(256) void extract_kernel(
    const float* __restrict__ x_full, float* __restrict__ x, long long n16) {
  long long t = (long long)blockIdx.x * blockDim.x + threadIdx.x;
  if (t >= n16) return;
  long long i = t >> 4;
  int j = (int)(t & 15);
  x[i * NFD + j] = x_full[i * XCOLS + 4 + j];
}

// ---------------- SpMM: hout[col] += norm[e] * hin[row] ----------------
// One thread handles one (edge, 4-feature chunk). Row data is contiguous
// (64B or 512B per node row), node matrix fits in the 192MB L2, so gathers
// and f32 atomic scatters stay on-chip.

__global__ __launch_bounds__(256) void spmm_scatter(
    const float* __restrict__ hin, float* __restrict__ hout,
    const float* __restrict__ nrm, const long long* __restrict__ ei,
    long long E, int F) {
  const long long chunks = F >> 2;
  long long t = (long long)blockIdx.x * blockDim.x + threadIdx.x;
  long long total = 2 * E * chunks;
  if (t >= total) return;
  long long e = t / chunks;
  int c4 = (int)(t - e * chunks) << 2;
  long long r, c;
  if (e < E) { r = ei[e];           c = ei[E + e]; }
  else       { r = ei[E + (e - E)]; c = ei[e - E]; }
  float s = nrm[e];
  const float4 v = *(const float4*)(hin + r * F + c4);
  float* dst = hout + c * F + c4;
  atomicAdd(dst + 0, v.x * s);
  atomicAdd(dst + 1, v.y * s);
  atomicAdd(dst + 2, v.z * s);
  atomicAdd(dst + 3, v.w * s);
}

// ---------------- fused 4-way GEMM + bias (+ReLU) via f32 WMMA ----------------
// out[N,H] = act( h0@W[0] + h1@W[1] + h2@W[2] + h3@W[3] + b ), W layout (4,F,H).
// Wave-per-16x16-tile; V_WMMA_F32_16X16X4_F32 accumulation, K stepped by 4.

__global__ __launch_bounds__(256) void tag_fused_gemm(
    const float* __restrict__ h0, const float* __restrict__ h1,
    const float* __restrict__ h2, const float* __restrict__ h3,
    const float* __restrict__ W, const float* __restrict__ bias,
    float* __restrict__ out, int Nn, int F, int H, int do_relu) {
  const int lane = threadIdx.x & 31;
  const int wave = (int)(((long long)blockIdx.x * blockDim.x + threadIdx.x) >> 5);
  const int tilesN = H >> 4;
  const int tilesM = Nn >> 4;
  if (wave >= tilesM * tilesN) return;           // wave-uniform: EXEC stays all-1s
  const int mt = wave / tilesN;
  const int nt = wave - mt * tilesN;
  const int lh = lane >> 4;                       // K-pair select (ISA 7.12.2)
  const int lm = lane & 15;
  const long long arow = (long long)(mt * 16 + lm);
  const int bcol = nt * 16 + lm;

  v8f c = {0.f, 0.f, 0.f, 0.f, 0.f, 0.f, 0.f, 0.f};

  const float* hs[4] = {h0, h1, h2, h3};
#pragma unroll
  for (int mat = 0; mat < 4; ++mat) {
    const float* __restrict__ hp = hs[mat] + arow * F;
    const float* __restrict__ wp = W + (size_t)mat * F * H + bcol;
    for (int k = 0; k < F; k += 4) {
      // A 16x4: lane lm = row M, v0/v1 = K = 2*lh, 2*lh+1 (contiguous 8B load)
      v2f a = *(const v2f*)(hp + k + 2 * lh);
      // B 4x16: lane lm = col N, v0/v1 = rows K = 2*lh, 2*lh+1
      v2f b;
      b.x = wp[(size_t)(k + 2 * lh) * H];
      b.y = wp[(size_t)(k + 2 * lh + 1) * H];
      c = __builtin_amdgcn_wmma_f32_16x16x4_f32(
          /*neg_a=*/false, a, /*neg_b=*/false, b,
          /*c_mod=*/(short)0, c, /*reuse_a=*/false, /*reuse_b=*/false);
    }
  }

  const float bb = bias[bcol];
#pragma unroll
  for (int r = 0; r < 8; ++r) {                   // C/D: M = r + 8*lh, N = lm
    float v = c[r] + bb;
    if (do_relu) v = fmaxf(v, 0.0f);
    out[(long long)(mt * 16 + r + 8 * lh) * H + bcol] = v;
  }
}

// ---------------- host orchestration ----------------

static inline int blocks_for(long long n, int bs) { return (int)((n + bs - 1) / bs); }

extern "C" void kernel_launch(void* const* d_in, const int* in_sizes, int n_in,
                              void* d_out, int out_size, void* d_ws, size_t ws_size,
                              hipStream_t stream) {
  const float*     x_full = (const float*)d_in[0];
  const long long* ei     = (const long long*)d_in[1];
  // d_in[2] edge_attr: unused by the reference math
  const float* w0 = (const float*)d_in[3];  const float* b0 = (const float*)d_in[4];
  const float* w1 = (const float*)d_in[5];  const float* b1 = (const float*)d_in[6];
  const float* w2 = (const float*)d_in[7];  const float* b2 = (const float*)d_in[8];
  const float* w3 = (const float*)d_in[9];  const float* b3 = (const float*)d_in[10];

  const int       Nn = in_sizes[0] / XCOLS;          // 100000 (divisible by 16)
  const long long E  = (long long)in_sizes[1] / 2;   // 1.6M

  // workspace carve-out (256B aligned slices)
  char* ws = (char*)d_ws;
  size_t off = 0;
  auto carve = [&](size_t bytes) -> float* {
    float* p = (float*)(ws + off);
    off += (bytes + 255) & ~(size_t)255;
    return p;
  };
  float* nrm = carve((size_t)2 * E * sizeof(float));
  float* deg = carve((size_t)Nn * sizeof(float));
  float* dis = carve((size_t)Nn * sizeof(float));
  float* xA  = carve((size_t)Nn * HD * sizeof(float));
  float* xB  = carve((size_t)Nn * HD * sizeof(float));
  float* h1  = carve((size_t)Nn * HD * sizeof(float));
  float* h2  = carve((size_t)Nn * HD * sizeof(float));
  float* h3  = carve((size_t)Nn * HD * sizeof(float));

  const int BS = 256;

  // ---- normalization coefficients ----
  hipMemsetAsync(deg, 0, (size_t)Nn * sizeof(float), stream);
  degree_kernel<<<blocks_for(E, BS), BS, 0, stream>>>(ei, deg, E);
  dis_kernel<<<blocks_for(Nn, BS), BS, 0, stream>>>(deg, dis, Nn);
  norm_kernel<<<blocks_for(2 * E, BS), BS, 0, stream>>>(ei, dis, nrm, E);

  // ---- x = x_full[:, 4:20] ----
  extract_kernel<<<blocks_for((long long)Nn * NFD, BS), BS, 0, stream>>>(
      x_full, xA, (long long)Nn * NFD);

  // ---- one TAGConv layer ----
  auto run_layer = [&](const float* xin, float* xout, int F, int H,
                       const float* W, const float* bias, int relu) {
    const size_t fbytes = (size_t)Nn * F * sizeof(float);
    const long long spmm_threads = 2 * E * (F >> 2);
    hipMemsetAsync(h1, 0, fbytes, stream);
    spmm_scatter<<<blocks_for(spmm_threads, BS), BS, 0, stream>>>(xin, h1, nrm, ei, E, F);
    hipMemsetAsync(h2, 0, fbytes, stream);
    spmm_scatter<<<blocks_for(spmm_threads, BS), BS, 0, stream>>>(h1, h2, nrm, ei, E, F);
    hipMemsetAsync(h3, 0, fbytes, stream);
    spmm_scatter<<<blocks_for(spmm_threads, BS), BS, 0, stream>>>(h2, h3, nrm, ei, E, F);
    const long long waves = (long long)(Nn >> 4) * (H >> 4);
    tag_fused_gemm<<<blocks_for(waves * 32, BS), BS, 0, stream>>>(
        xin, h1, h2, h3, W, bias, xout, Nn, F, H, relu);
  };

  run_layer(xA, xB, NFD, HD, w0, b0, 1);
  run_layer(xB, xA, HD,  HD, w1, b1, 1);
  run_layer(xA, xB, HD,  HD, w2, b2, 1);
  run_layer(xB, (float*)d_out, HD, NFD, w3, b3, 0);
}